// Lateral_309237646072
// MI455X (gfx1250) — compile-verified
//
#include <hip/hip_runtime.h>

// Lateral affine combine: y[0] = x[0]; y[i] = x[i] + w[i-1]*x[i-1] + b[i-1]
// H = D = 8192, fp32. Pure HBM-streaming kernel: ~512 MB traffic, ~22 us floor
// at 23.3 TB/s. Intentionally no WMMA: arithmetic intensity is 0.25 FLOP/byte,
// so matrix ops cannot change the roofline; the win is 128-bit loads, scalar
// (wave-uniform) weight/bias, and non-temporal 128-bit stores so the y stream
// does not evict x rows that still have one pending reuse in L2.

typedef float v4f __attribute__((ext_vector_type(4)));

__global__ __launch_bounds__(256) void lateral_affine_kernel(
    const float* __restrict__ x,
    const float* __restrict__ w,
    const float* __restrict__ b,
    float* __restrict__ y,
    int d)  // row width in floats (8192)
{
    const int row = blockIdx.y;                 // 0..H-1 (uniform per block)
    const size_t rowBase = (size_t)row * (size_t)d;

    const v4f* xr = (const v4f*)(x + rowBase);
    v4f*       yr = (v4f*)(y + rowBase);

    // Each block covers a tile of 512 float4s (2048 floats); each thread owns
    // two float4 chunks 256 apart -> two independent b128 loads in flight.
    const int i0 = blockIdx.x * 512 + threadIdx.x;
    const int i1 = i0 + 256;

    v4f a0 = xr[i0];
    v4f a1 = xr[i1];

    if (row == 0) {
        // First row passes through unchanged.
        __builtin_nontemporal_store(a0, &yr[i0]);
        __builtin_nontemporal_store(a1, &yr[i1]);
        return;
    }

    // Scalar per-row weight/bias: block-uniform -> scalarizable loads.
    const float wv = w[row - 1];
    const float bv = b[row - 1];

    const v4f* xp = (const v4f*)(x + rowBase - (size_t)d);
    v4f p0 = xp[i0];
    v4f p1 = xp[i1];

    v4f o0, o1;
#pragma unroll
    for (int k = 0; k < 4; ++k) {
        o0[k] = __builtin_fmaf(p0[k], wv, a0[k] + bv);
        o1[k] = __builtin_fmaf(p1[k], wv, a1[k] + bv);
    }

    // y is write-once / never re-read: non-temporal b128 stores keep it from
    // thrashing L2 (which is caching the x-row reuse window).
    __builtin_nontemporal_store(o0, &yr[i0]);
    __builtin_nontemporal_store(o1, &yr[i1]);
}

extern "C" void kernel_launch(void* const* d_in, const int* in_sizes, int n_in,
                              void* d_out, int out_size, void* d_ws, size_t ws_size,
                              hipStream_t stream) {
    const float* x = (const float*)d_in[0];   // [H, D] fp32
    const float* w = (const float*)d_in[1];   // [H-1, 1] fp32
    const float* b = (const float*)d_in[2];   // [H-1, 1] fp32
    float* y = (float*)d_out;                 // [H, D] fp32

    const int Hm1 = in_sizes[1];              // H-1 = 8191
    const int H   = Hm1 + 1;                  // 8192
    const int D   = in_sizes[0] / H;          // 8192

    dim3 block(256);                          // 8 wave32 waves per block
    dim3 grid((unsigned)(D / 2048), (unsigned)H);  // (4, 8192)
    lateral_affine_kernel<<<grid, block, 0, stream>>>(x, w, b, y, D);
}